// MoEGate_77558519431578
// MI455X (gfx1250) — compile-verified
//
#include <hip/hip_runtime.h>
#include <math.h>

typedef __attribute__((ext_vector_type(2))) float v2f;
typedef __attribute__((ext_vector_type(8))) float v8f;

#define T_TOKENS 16384
#define HDIM     2048
#define NEXP     64
#define TOPK     8

// Block = 128 threads = 4 waves (wave32). Each wave computes a 16-token x 64-expert
// logit tile with V_WMMA_F32_16X16X4_F32, then lanes 0-15 do per-token top-8.
__global__ __launch_bounds__(128) void moe_gate_kernel(
    const float* __restrict__ x,      // [T, H]
    const float* __restrict__ w,      // [E, H]
    const float* __restrict__ bias,   // [E]
    float* __restrict__ out_idx,      // [T, 8] indices stored as float
    float* __restrict__ out_w)        // [T, 8] normalized weights
{
    __shared__ float lds_logits[4][16][NEXP + 1];   // +1 pad -> conflict-free column reads
    __shared__ float lds_bias[NEXP];

    const int lane = threadIdx.x & 31;
    const int wave = threadIdx.x >> 5;
    const int tile = blockIdx.x * 4 + wave;         // 16-token tile id
    const int t0   = tile * 16;

    if (threadIdx.x < NEXP) lds_bias[threadIdx.x] = bias[threadIdx.x];

    // A fragment addressing (16x4 f32): lanes 0-15 -> K={0,1}, lanes 16-31 -> K={2,3},
    // M = lane % 16. B mirrors this with N = lane % 16.
    const int mn    = lane & 15;
    const int khalf = (lane >> 4) * 2;

    const float* xrow = x + (size_t)(t0 + mn) * HDIM + khalf;
    const float* wr0  = w + (size_t)(mn +  0) * HDIM + khalf;
    const float* wr1  = w + (size_t)(mn + 16) * HDIM + khalf;
    const float* wr2  = w + (size_t)(mn + 32) * HDIM + khalf;
    const float* wr3  = w + (size_t)(mn + 48) * HDIM + khalf;

    v8f acc0 = {}, acc1 = {}, acc2 = {}, acc3 = {};

#pragma unroll 4
    for (int k = 0; k < HDIM; k += 4) {
        v2f a  = *(const v2f*)(xrow + k);
        v2f b0 = *(const v2f*)(wr0 + k);
        v2f b1 = *(const v2f*)(wr1 + k);
        v2f b2 = *(const v2f*)(wr2 + k);
        v2f b3 = *(const v2f*)(wr3 + k);
        // 8 args: (neg_a, A, neg_b, B, c_mod, C, reuse_a, reuse_b)
        acc0 = __builtin_amdgcn_wmma_f32_16x16x4_f32(false, a, false, b0, (short)0, acc0, false, false);
        acc1 = __builtin_amdgcn_wmma_f32_16x16x4_f32(false, a, false, b1, (short)0, acc1, false, false);
        acc2 = __builtin_amdgcn_wmma_f32_16x16x4_f32(false, a, false, b2, (short)0, acc2, false, false);
        acc3 = __builtin_amdgcn_wmma_f32_16x16x4_f32(false, a, false, b3, (short)0, acc3, false, false);
    }

    // C/D layout: lanes 0-15: VGPR r -> M=r, N=lane; lanes 16-31: VGPR r -> M=8+r, N=lane-16.
    const int mrow = (lane >> 4) * 8;
    const int ncol = lane & 15;
#pragma unroll
    for (int r = 0; r < 8; ++r) {
        lds_logits[wave][mrow + r][ncol +  0] = acc0[r];
        lds_logits[wave][mrow + r][ncol + 16] = acc1[r];
        lds_logits[wave][mrow + r][ncol + 32] = acc2[r];
        lds_logits[wave][mrow + r][ncol + 48] = acc3[r];
    }
    __syncthreads();

    if (lane < 16) {
        const int t = t0 + lane;
        const float* row = &lds_logits[wave][lane][0];

        unsigned long long used = 0ull;
        int   isel[TOPK];
        float psel[TOPK];
        float psum = 0.0f;

#pragma unroll
        for (int j = 0; j < TOPK; ++j) {
            float best = -INFINITY;
            int   bi   = 0;
            for (int e = 0; e < NEXP; ++e) {
                if (used & (1ull << e)) continue;
                float v = row[e] + lds_bias[e];     // biased logit for selection
                if (v > best) { best = v; bi = e; } // strict > : ties -> lowest index (jax top_k)
            }
            used |= (1ull << bi);
            float g = row[bi];
            float p = 1.0f / (1.0f + expf(-g));     // sigmoid of raw logit
            isel[j] = bi;
            psel[j] = p;
            psum   += p;
        }

        const float denom = fmaxf(psum, 1e-12f);
#pragma unroll
        for (int j = 0; j < TOPK; ++j) {
            out_idx[(size_t)t * TOPK + j] = (float)isel[j];
            out_w  [(size_t)t * TOPK + j] = psel[j] / denom;
        }
    }
}

extern "C" void kernel_launch(void* const* d_in, const int* in_sizes, int n_in,
                              void* d_out, int out_size, void* d_ws, size_t ws_size,
                              hipStream_t stream) {
    (void)in_sizes; (void)n_in; (void)out_size; (void)d_ws; (void)ws_size;
    const float* x    = (const float*)d_in[0];   // hidden_states [4,4096,2048]
    const float* w    = (const float*)d_in[1];   // weight [64,2048]
    const float* bias = (const float*)d_in[2];   // expert_biases [64]

    float* out      = (float*)d_out;
    float* out_idx  = out;                       // [4,4096,8] indices (as float)
    float* out_wt   = out + (size_t)T_TOKENS * TOPK;  // [4,4096,8] weights

    dim3 grid(T_TOKENS / 64);   // 4 waves/block * 16 tokens/wave
    dim3 block(128);
    hipLaunchKernelGGL(moe_gate_kernel, grid, block, 0, stream,
                       x, w, bias, out_idx, out_wt);
}